// ScaleAwareRouter_88527865905617
// MI455X (gfx1250) — compile-verified
//
#include <hip/hip_runtime.h>
#include <math.h>
#include <stdint.h>

typedef __attribute__((ext_vector_type(2))) float        v2f;
typedef __attribute__((ext_vector_type(8))) float        v8f;
typedef __attribute__((ext_vector_type(4))) unsigned int v4u;
typedef __attribute__((ext_vector_type(8))) int          v8i;
typedef __attribute__((ext_vector_type(4))) int          v4i;

#define INPUT_DIM   2048
#define NUM_EXPERTS 64
#define KC          32                 // K chunk per TDM tile
#define ROWS        64                 // tokens per workgroup
#define LSTR        36                 // KC + 4 dword TDM pad -> conflict-free b64 reads
#define NTHREADS    128                // 4 wave32s
#define NCHUNK      (INPUT_DIM / KC)   // 64
#define TILE_F      (ROWS * LSTR)      // 2304 floats per LDS tile (x and W tiles same shape)
#define TEMP_INV    (1.0f / 0.7f)
#define JITTER      0.01f

// ---------------------------------------------------------------------------
// TDM: 2D tile load global -> LDS with padding (pad 4 dwords after every 32),
// per CDNA5 ISA D# layout. data_size=4B, type=2 ("image").
// ---------------------------------------------------------------------------
__device__ __forceinline__ void tdm_load_2d(unsigned lds_byte_off,
                                            const float* gaddr,
                                            unsigned tensor_d0, unsigned tensor_d1,
                                            unsigned tile_d0,   unsigned tile_d1,
                                            unsigned stride0_elems)
{
    const unsigned long long ga = (unsigned long long)(uintptr_t)gaddr;
    v4u g0;
    g0[0] = 1u;                                               // count=1, user descriptor
    g0[1] = lds_byte_off;                                     // lds_addr (bytes)
    g0[2] = (unsigned)ga;                                     // global_addr[31:0]
    g0[3] = (unsigned)((ga >> 32) & 0x01FFFFFFu) | (2u << 30);// global_addr[56:32] | type=2
    v8i g1;
    g1[0] = (int)((2u << 16)      // data_size = 4 bytes
                | (1u << 20)      // pad_enable
                | (5u << 22)      // pad_interval: 2^(5+1)=64 x 8B? -> 32 dwords (see note)
                | (3u << 25));    // pad_amount: 4 dwords
    // pad_interval encoding is in dwords {0=2,1=4,...}: 32 dwords -> 4; bytesStored>>3
    // check in ISA pseudocode counts 8B units: 32 dwords = 16 units -> 1<<4. Use 4.
    g1[0] = (int)((2u << 16) | (1u << 20) | (4u << 22) | (3u << 25));
    g1[1] = (int)((tensor_d0 & 0xFFFFu) << 16);                       // barrier_addr=0 | dim0.lo
    g1[2] = (int)((tensor_d0 >> 16) | ((tensor_d1 & 0xFFFFu) << 16)); // dim0.hi | dim1.lo
    g1[3] = (int)((tensor_d1 >> 16) | ((tile_d0 & 0xFFFFu) << 16));   // dim1.hi | tile_dim0
    g1[4] = (int)(tile_d1 & 0xFFFFu);                                 // tile_dim1 | tile_dim2=0
    g1[5] = (int)stride0_elems;                                       // tensor_dim0_stride lo
    g1[6] = 0;                                                        // stride hi | dim1_stride lo
    g1[7] = 0;
    const v4i z4 = {0, 0, 0, 0};
#if __has_include(<hip/amd_detail/amd_gfx1250_TDM.h>)
    const v8i z8 = {0, 0, 0, 0, 0, 0, 0, 0};
    __builtin_amdgcn_tensor_load_to_lds(g0, g1, z4, z4, z8, 0);       // clang-23 6-arg form
#else
    __builtin_amdgcn_tensor_load_to_lds(g0, g1, z4, z4, 0);           // ROCm 7.2 5-arg form
#endif
}

// Deterministic hash -> approx standard normal (Box-Muller); threefry stream
// is not reproducible here, jitter is 1% noise on logits.
__device__ __forceinline__ float gauss_noise(unsigned a, unsigned b) {
    unsigned h = (a ^ 0x9E3779B9u) * 0x85EBCA6Bu;
    h ^= b + 0x165667B1u;  h *= 0xC2B2AE35u;
    h ^= h >> 16;          h *= 0x27D4EB2Fu;  h ^= h >> 13;
    unsigned h2 = h * 0x9E3779B1u + (b * 0x6C8E9CF5u + 0x94D049BBu);
    h2 ^= h2 >> 17;
    float u1 = ((h  & 0xFFFFFFu) + 1u) * (1.0f / 16777217.0f);
    float u2 =  (h2 & 0xFFFFFFu)       * (1.0f / 16777216.0f);
    return sqrtf(-2.0f * __logf(u1)) * __cosf(6.28318530718f * u2);
}

// ---------------------------------------------------------------------------
// Kernel 1: sbias[e] = b[e] + dot(scale_emb[sidx], W[e])  (folds the scale
// conditioning out of the GEMM: (x+s)Wt + b = xWt + (sWt + b)).
// ---------------------------------------------------------------------------
__global__ __launch_bounds__(256)
void router_sbias_kernel(const float* __restrict__ semb,
                         const float* __restrict__ W,
                         const float* __restrict__ bias,
                         const int*   __restrict__ sidx_p,
                         float*       __restrict__ sbias)
{
    __shared__ float part[256];
    const int tid = threadIdx.x;
    const int e   = tid & 63;
    const int seg = tid >> 6;                 // 4 segments x 512 K each
    const int sidx = *sidx_p;
    float acc = 0.0f;
    if (sidx >= 0) {
        const float* s = semb + (size_t)sidx * INPUT_DIM + seg * 512;
        const float* w = W    + (size_t)e    * INPUT_DIM + seg * 512;
        for (int k = 0; k < 512; k += 4) {
            const float4 sv = *reinterpret_cast<const float4*>(s + k);
            const float4 wv = *reinterpret_cast<const float4*>(w + k);
            acc += sv.x * wv.x + sv.y * wv.y + sv.z * wv.z + sv.w * wv.w;
        }
    }
    part[tid] = acc;
    __syncthreads();
    if (tid < 64)
        sbias[tid] = bias[tid] + part[tid] + part[tid + 64] + part[tid + 128] + part[tid + 192];
}

// ---------------------------------------------------------------------------
// Kernel 2: x @ Wt via f32 WMMA, TDM double-buffered LDS staging, fused
// jitter/temperature/top-2/softmax epilogue.
// ---------------------------------------------------------------------------
__global__ __launch_bounds__(NTHREADS)
void ScaleAwareRouter_wmma_tdm_kernel(const float* __restrict__ x,
                                      const float* __restrict__ W,
                                      const float* __restrict__ sbias,
                                      float*       __restrict__ out_w,
                                      int*         __restrict__ out_i,
                                      int M)
{
    // [0..1] x tiles, [2..3] W tiles; logits reuse the x double-buffer region.
    __shared__ float lds[4 * TILE_F];

    const int tid  = threadIdx.x;
    const int wave = tid >> 5;
    const int lane = tid & 31;
    const int m0   = blockIdx.x * ROWS;
    const unsigned lds_base = (unsigned)(uintptr_t)(void*)lds;  // flat low bits = LDS offset

    v8f acc0 = {}, acc1 = {}, acc2 = {}, acc3 = {};

    // A fragment (16x4 f32): lanes 0-15 rows, k+0/k+1; lanes 16-31 rows, k+2/k+3.
    const int arow = wave * 16 + (lane & 15);
    const int koff = (lane >> 4) << 1;
    const int brow = lane & 15;

    const float* xrow = x + (size_t)m0 * INPUT_DIM;

    // Prologue: wave 0 issues TDM DMA for chunk 0 (EXEC is ignored by tensor ops).
    if (tid < 32) {
        tdm_load_2d(lds_base + 0 * TILE_F * 4, xrow, INPUT_DIM, (unsigned)M,  KC, ROWS,        INPUT_DIM);
        tdm_load_2d(lds_base + 2 * TILE_F * 4, W,    INPUT_DIM, NUM_EXPERTS,  KC, NUM_EXPERTS, INPUT_DIM);
    }

    for (int c = 0; c < NCHUNK; ++c) {
        const int b = c & 1;
        __syncthreads();                 // buffer b^1 is free (chunk c-1 consumed)
        if (tid < 32) {
            if (c + 1 < NCHUNK) {
                const int k0n = (c + 1) * KC;
                tdm_load_2d(lds_base + (unsigned)(b ^ 1) * TILE_F * 4,       xrow + k0n,
                            INPUT_DIM, (unsigned)M, KC, ROWS, INPUT_DIM);
                tdm_load_2d(lds_base + (unsigned)(2 + (b ^ 1)) * TILE_F * 4, W + k0n,
                            INPUT_DIM, NUM_EXPERTS, KC, NUM_EXPERTS, INPUT_DIM);
                __builtin_amdgcn_s_wait_tensorcnt(2);   // chunk c landed, c+1 in flight
            } else {
                __builtin_amdgcn_s_wait_tensorcnt(0);
            }
        }
        __syncthreads();                 // publish chunk c to all waves

        const float* xb  = &lds[b * TILE_F];
        const float* wb  = &lds[(2 + b) * TILE_F];
        const float* ax  = &xb[arow * LSTR + koff];
        const float* b0p = &wb[( 0 + brow) * LSTR + koff];
        const float* b1p = &wb[(16 + brow) * LSTR + koff];
        const float* b2p = &wb[(32 + brow) * LSTR + koff];
        const float* b3p = &wb[(48 + brow) * LSTR + koff];
#pragma unroll
        for (int kk = 0; kk < KC; kk += 4) {
            const v2f a  = *reinterpret_cast<const v2f*>(ax  + kk);
            const v2f b0 = *reinterpret_cast<const v2f*>(b0p + kk);
            const v2f b1 = *reinterpret_cast<const v2f*>(b1p + kk);
            const v2f b2 = *reinterpret_cast<const v2f*>(b2p + kk);
            const v2f b3 = *reinterpret_cast<const v2f*>(b3p + kk);
            acc0 = __builtin_amdgcn_wmma_f32_16x16x4_f32(false, a, false, b0, (short)0, acc0, false, false);
            acc1 = __builtin_amdgcn_wmma_f32_16x16x4_f32(false, a, false, b1, (short)0, acc1, false, false);
            acc2 = __builtin_amdgcn_wmma_f32_16x16x4_f32(false, a, false, b2, (short)0, acc2, false, false);
            acc3 = __builtin_amdgcn_wmma_f32_16x16x4_f32(false, a, false, b3, (short)0, acc3, false, false);
        }
    }

    // ---- dump logits to LDS (C/D layout: VGPR i, lanes 0-15 -> M=i, 16-31 -> M=i+8)
    float* ls = lds;                            // 64 x 65 floats fits in x buffers (4608 f)
    const int trow = wave * 16 + 8 * (lane >> 4);
    const int ecol = lane & 15;
#pragma unroll
    for (int i = 0; i < 8; ++i) {
        ls[(trow + i) * 65 +  0 + ecol] = acc0[i];
        ls[(trow + i) * 65 + 16 + ecol] = acc1[i];
        ls[(trow + i) * 65 + 32 + ecol] = acc2[i];
        ls[(trow + i) * 65 + 48 + ecol] = acc3[i];
    }
    __syncthreads();

    // ---- per-token epilogue: sbias + jitter + temperature, stable top-2, softmax(2)
    if (tid < ROWS) {
        const int gm = m0 + tid;
        float v1 = -3.4e38f, v2 = -3.4e38f;
        int   i1 = 0, i2 = 0;
        for (int e = 0; e < NUM_EXPERTS; ++e) {
            const float v = (ls[tid * 65 + e] + sbias[e] +
                             JITTER * gauss_noise((unsigned)gm, (unsigned)e)) * TEMP_INV;
            if (v > v1)      { v2 = v1; i2 = i1; v1 = v; i1 = e; }
            else if (v > v2) { v2 = v;  i2 = e; }
        }
        const float d   = __expf(v2 - v1);
        const float inv = 1.0f / (1.0f + d);
        out_w[gm * 2 + 0] = inv;
        out_w[gm * 2 + 1] = d * inv;
        out_i[gm * 2 + 0] = i1;
        out_i[gm * 2 + 1] = i2;
    }
}

extern "C" void kernel_launch(void* const* d_in, const int* in_sizes, int n_in,
                              void* d_out, int out_size, void* d_ws, size_t ws_size,
                              hipStream_t stream) {
    const float* x    = (const float*)d_in[0];
    const float* semb = (const float*)d_in[1];
    const float* W    = (const float*)d_in[2];
    const float* bias = (const float*)d_in[3];
    const int*   sidx = (const int*)d_in[4];
    const int M = in_sizes[0] / INPUT_DIM;            // 16384 tokens
    float* out_w = (float*)d_out;                     // [M,2] weights
    int*   out_i = (int*)d_out + (out_size / 2);      // [M,2] indices (i32 region)
    float* sbias = (float*)d_ws;                      // 64 floats of scratch

    router_sbias_kernel<<<dim3(1), dim3(256), 0, stream>>>(semb, W, bias, sidx, sbias);
    ScaleAwareRouter_wmma_tdm_kernel<<<dim3(M / ROWS), dim3(NTHREADS), 0, stream>>>(
        x, W, sbias, out_w, out_i, M);
}